// GAT_37976100831375
// MI455X (gfx1250) — compile-verified
//
#include <hip/hip_runtime.h>
#include <math.h>

typedef __attribute__((ext_vector_type(2))) float v2f;
typedef __attribute__((ext_vector_type(8))) float v8f;

#define GAT_NGRAPH 256
#define GAT_FEAT   64
#define GAT_KMAX   72

// ---------------------------------------------------------------------------
// float atomic max via monotone int/uint mapping (init must be -inf)
// ---------------------------------------------------------------------------
__device__ __forceinline__ void atomic_max_f32(float* addr, float val) {
  if (val >= 0.0f) atomicMax((int*)addr, __float_as_int(val));
  else             atomicMin((unsigned int*)addr, __float_as_uint(val));
}

// ---------------------------------------------------------------------------
// H[N,64] = X[N,K] @ W[K,64]  (fp32, V_WMMA_F32_16X16X4_F32)
// One wave32 -> 16 rows x 64 cols (4 accumulators), K-loop in steps of 4.
// W (<=18KB) is staged into LDS once per block and shared by all 8 waves,
// so the inner loop's VMEM pipe only streams A rows of X.
// Fragment layouts per CDNA5 ISA 7.12.2:
//   A 16x4 : VGPR0 = {K=0 lanes0-15, K=2 lanes16-31}, VGPR1 = {K=1, K=3}
//   B 4x16 : VGPR0 = {K=0 lanes0-15, K=2 lanes16-31}, VGPR1 = {K=1, K=3}, N=lane&15
//   C 16x16: VGPR v = row (v + 8*(lane>=16)), col = lane&15
// ---------------------------------------------------------------------------
__global__ __launch_bounds__(256)
void gat_gemm64(const float* __restrict__ X, const float* __restrict__ W,
                float* __restrict__ H, int N, int K) {
  __shared__ float wlds[GAT_KMAX * GAT_FEAT];

  const int lane = threadIdx.x;        // 0..31
  const int wave = threadIdx.y;        // 0..7
  const int tid  = wave * 32 + lane;

  // cooperative stage of W into LDS (all waves participate before barrier)
  for (int i = tid; i < K * GAT_FEAT; i += 256) wlds[i] = W[i];
  __syncthreads();

  const int r0 = (blockIdx.x * 8 + wave) * 16;
  if (r0 >= N) return;                 // wave-uniform: EXEC all-ones for WMMA
  const int m  = lane & 15;
  const int hi = lane >> 4;            // 0 or 1
  int row = r0 + m; if (row > N - 1) row = N - 1;   // clamp loads for ragged tail
  const float* xr = X + (size_t)row * K;

  v8f c0 = {}, c1 = {}, c2 = {}, c3 = {};
  for (int kk = 0; kk < K; kk += 4) {
    v2f a;
    a.x = xr[kk + 2 * hi + 0];
    a.y = xr[kk + 2 * hi + 1];
    const float* w0 = wlds + (kk + 2 * hi) * GAT_FEAT + m;
    const float* w1 = w0 + GAT_FEAT;
    v2f b;
    b.x = w0[0];  b.y = w1[0];
    c0 = __builtin_amdgcn_wmma_f32_16x16x4_f32(false, a, false, b, (short)0, c0, false, false);
    b.x = w0[16]; b.y = w1[16];
    c1 = __builtin_amdgcn_wmma_f32_16x16x4_f32(false, a, false, b, (short)0, c1, false, false);
    b.x = w0[32]; b.y = w1[32];
    c2 = __builtin_amdgcn_wmma_f32_16x16x4_f32(false, a, false, b, (short)0, c2, false, false);
    b.x = w0[48]; b.y = w1[48];
    c3 = __builtin_amdgcn_wmma_f32_16x16x4_f32(false, a, false, b, (short)0, c3, false, false);
  }
#pragma unroll
  for (int v = 0; v < 8; ++v) {
    int rr = r0 + v + 8 * hi;
    if (rr < N) {
      float* hp = H + (size_t)rr * GAT_FEAT + m;
      hp[0]  = c0[v];
      hp[16] = c1[v];
      hp[32] = c2[v];
      hp[48] = c3[v];
    }
  }
}

// per-node attention logits: als[i] = h[i,:].a_src ; ald[i] = h[i,:].a_dst
__global__ void gat_node_dots(const float* __restrict__ H,
                              const float* __restrict__ asrc,
                              const float* __restrict__ adst,
                              float* __restrict__ als, float* __restrict__ ald, int N) {
  int i = blockIdx.x * blockDim.x + threadIdx.x;
  if (i >= N) return;
  const float* h = H + (size_t)i * GAT_FEAT;
  float s1 = 0.f, s2 = 0.f;
#pragma unroll 8
  for (int k = 0; k < GAT_FEAT; ++k) { float v = h[k]; s1 += v * asrc[k]; s2 += v * adst[k]; }
  als[i] = s1; ald[i] = s2;
}

__global__ void gat_fill(float* __restrict__ p, float v, int n) {
  int i = blockIdx.x * blockDim.x + threadIdx.x;
  if (i < n) p[i] = v;
}

__device__ __forceinline__ void gat_edge_sd(const int* ei, int NE, int e, int& s, int& d) {
  if (e < NE) { s = ei[e]; d = ei[NE + e]; }
  else        { s = e - NE; d = e - NE; }      // appended self-loops
}

// pass 1: leaky logits + segment max into mx[dst]
__global__ void gat_edge_logits(const int* __restrict__ ei, int NE, int N,
                                const float* __restrict__ als, const float* __restrict__ ald,
                                float* __restrict__ elog, float* __restrict__ mx) {
  int e = blockIdx.x * blockDim.x + threadIdx.x;
  int ET = NE + N;
  if (e >= ET) return;
  int s, d; gat_edge_sd(ei, NE, e, s, d);
  float x = als[s] + ald[d];
  float lg = (x >= 0.f) ? x : 0.2f * x;        // GAT negative slope
  elog[e] = lg;
  atomic_max_f32(mx + d, lg);
}

// pass 2: ex = exp(logit - mx[dst]) (in place), segment sum into sm[dst]
__global__ void gat_edge_exp(const int* __restrict__ ei, int NE, int N,
                             float* __restrict__ elog,
                             const float* __restrict__ mx, float* __restrict__ sm) {
  int e = blockIdx.x * blockDim.x + threadIdx.x;
  int ET = NE + N;
  if (e >= ET) return;
  int s, d; gat_edge_sd(ei, NE, e, s, d);
  float ex = __expf(elog[e] - mx[d]);
  elog[e] = ex;
  atomicAdd(sm + d, ex);
}

// pass 3: out[dst,:] += alpha * h[src,:]  (one wave per edge, 2 feats/lane)
__global__ __launch_bounds__(256)
void gat_edge_agg(const int* __restrict__ ei, int NE, int N,
                  const float* __restrict__ ex, const float* __restrict__ sm,
                  const float* __restrict__ H, float* __restrict__ OUT) {
  int lane = threadIdx.x;
  int e = blockIdx.x * 8 + threadIdx.y;
  int ET = NE + N;
  if (e >= ET) return;
  int s, d; gat_edge_sd(ei, NE, e, s, d);
  float alpha = ex[e] / (sm[d] + 1e-16f);
  const float* hs = H   + (size_t)s * GAT_FEAT + lane;
  float*       od = OUT + (size_t)d * GAT_FEAT + lane;
  atomicAdd(od,      hs[0]  * alpha);
  atomicAdd(od + 32, hs[32] * alpha);
}

// bias + (optional) leaky activation; slope==1.0f -> identity activation
__global__ void gat_bias_act(float* __restrict__ OUT, const float* __restrict__ b,
                             int N, float slope) {
  int idx = blockIdx.x * blockDim.x + threadIdx.x;
  if (idx >= N * GAT_FEAT) return;
  int k = idx & (GAT_FEAT - 1);
  float v = OUT[idx] + b[k];
  OUT[idx] = (v >= 0.f) ? v : slope * v;
}

// global mean pool, stage 1: per-graph sums + counts (one wave per node)
__global__ __launch_bounds__(256)
void gat_pool(const int* __restrict__ batch, const float* __restrict__ H,
              float* __restrict__ pool, float* __restrict__ cnt, int N) {
  int lane = threadIdx.x;
  int i = blockIdx.x * 8 + threadIdx.y;
  if (i >= N) return;
  int g = batch[i];
  const float* h = H + (size_t)i * GAT_FEAT;
  atomicAdd(pool + (size_t)g * GAT_FEAT + lane,      h[lane]);
  atomicAdd(pool + (size_t)g * GAT_FEAT + lane + 32, h[lane + 32]);
  if (lane == 0) atomicAdd(cnt + g, 1.0f);
}

// stage 2: out[g, j] = (sum[g,:]/max(cnt,1)) @ fcW[:, j] + fcb[j]
__global__ void gat_final_fc(const float* __restrict__ pool, const float* __restrict__ cnt,
                             const float* __restrict__ fcW, const float* __restrict__ fcb,
                             float* __restrict__ out) {
  int idx = blockIdx.x * blockDim.x + threadIdx.x;
  if (idx >= GAT_NGRAPH * 2) return;
  int g = idx >> 1, j = idx & 1;
  float c = cnt[g]; if (c < 1.f) c = 1.f;
  float acc = 0.f;
#pragma unroll 8
  for (int k = 0; k < GAT_FEAT; ++k) acc += pool[(size_t)g * GAT_FEAT + k] * fcW[k * 2 + j];
  out[idx] = acc / c + fcb[j];
}

// ---------------------------------------------------------------------------
extern "C" void kernel_launch(void* const* d_in, const int* in_sizes, int n_in,
                              void* d_out, int out_size, void* d_ws, size_t ws_size,
                              hipStream_t stream) {
  const float* x   = (const float*)d_in[0];
  const int*   ei  = (const int*)  d_in[1];
  const int*   bat = (const int*)  d_in[2];
  const float* W1  = (const float*)d_in[3];
  const float* as1 = (const float*)d_in[4];
  const float* ad1 = (const float*)d_in[5];
  const float* b1  = (const float*)d_in[6];
  const float* W2  = (const float*)d_in[7];
  const float* as2 = (const float*)d_in[8];
  const float* ad2 = (const float*)d_in[9];
  const float* b2  = (const float*)d_in[10];
  const float* fcW = (const float*)d_in[11];
  const float* fcb = (const float*)d_in[12];
  float* out = (float*)d_out;

  const int N  = in_sizes[0] / 72;      // 100000
  const int NE = in_sizes[1] / 2;       // 1600000
  const int ET = NE + N;                // edges + self-loops

  // workspace layout (floats); total ~60 MB
  float* p    = (float*)d_ws;
  float* hA   = p; p += (size_t)N * GAT_FEAT;   // h (GEMM output) per layer
  float* hB   = p; p += (size_t)N * GAT_FEAT;   // aggregated output per layer
  float* als  = p; p += N;
  float* ald  = p; p += N;
  float* mx   = p; p += N;
  float* sm   = p; p += N;
  float* elog = p; p += ET;                      // logits, then exp(e - m)
  float* pool = p; p += GAT_NGRAPH * GAT_FEAT;
  float* cnt  = p; p += GAT_NGRAPH;

  dim3 wb(32, 8);
  const int gemmBlocks = (N + 127) / 128;
  const int nodeB  = (N + 255) / 256;
  const int featB  = (N * GAT_FEAT + 255) / 256;
  const int edgeB  = (ET + 255) / 256;
  const int edgeWB = (ET + 7) / 8;

  auto layer = [&](const float* in, int K, const float* W, const float* av, const float* dv,
                   const float* bv, float* h, float* o, float slope) {
    gat_gemm64<<<gemmBlocks, wb, 0, stream>>>(in, W, h, N, K);
    gat_node_dots<<<nodeB, 256, 0, stream>>>(h, av, dv, als, ald, N);
    gat_fill<<<nodeB, 256, 0, stream>>>(mx, -INFINITY, N);
    gat_fill<<<nodeB, 256, 0, stream>>>(sm, 0.0f, N);
    gat_fill<<<featB, 256, 0, stream>>>(o, 0.0f, N * GAT_FEAT);
    gat_edge_logits<<<edgeB, 256, 0, stream>>>(ei, NE, N, als, ald, elog, mx);
    gat_edge_exp<<<edgeB, 256, 0, stream>>>(ei, NE, N, elog, mx, sm);
    gat_edge_agg<<<edgeWB, wb, 0, stream>>>(ei, NE, N, elog, sm, h, o);
    gat_bias_act<<<featB, 256, 0, stream>>>(o, bv, N, slope);
  };

  // layer 1: x[N,72] -> hB ; leaky 0.01 activation
  layer(x,  72, W1, as1, ad1, b1, hA, hB, 0.01f);
  // layer 2: hB[N,64] -> hB (gemm consumes hB into hA before hB is zeroed)
  layer(hB, 64, W2, as2, ad2, b2, hA, hB, 1.0f);

  // global mean pool + FC
  gat_fill<<<(GAT_NGRAPH * GAT_FEAT + 255) / 256, 256, 0, stream>>>(pool, 0.0f, GAT_NGRAPH * GAT_FEAT);
  gat_fill<<<1, 256, 0, stream>>>(cnt, 0.0f, GAT_NGRAPH);
  gat_pool<<<(N + 7) / 8, wb, 0, stream>>>(bat, hB, pool, cnt, N);
  gat_final_fc<<<2, 256, 0, stream>>>(pool, cnt, fcW, fcb, out);
}